// ChamferLoss_44581760532793
// MI455X (gfx1250) — compile-verified
//
#include <hip/hip_runtime.h>
#include <hip/hip_bf16.h>

typedef __attribute__((ext_vector_type(2))) float v2f;
typedef __attribute__((ext_vector_type(8))) float v8f;

// ---------------- constants (reference: B=8, N=M=4096, D=256) ----------------
constexpr int Dc      = 256;  // feature dim (K)
constexpr int BLK     = 128;  // output tile: 128 rows (gts) x 128 cols (preds)
constexpr int KC      = 16;   // K chunk staged in LDS (double buffered)
constexpr int LDS_STR = 20;   // padded row stride (floats): conflict-free banks
constexpr int NCHUNK  = Dc / KC;            // 16
constexpr int SEG     = BLK * LDS_STR;      // floats per LDS tile buffer

// Monotone float->uint key so unsigned atomicMin == float min (handles negatives).
__device__ __forceinline__ unsigned fkey(float f) {
    unsigned u = __float_as_uint(f);
    return (u & 0x80000000u) ? ~u : (u | 0x80000000u);
}
__device__ __forceinline__ float fdecode(unsigned k) {
    unsigned u = (k & 0x80000000u) ? (k & 0x7FFFFFFFu) : ~k;
    return __uint_as_float(u);
}

// ---------------- kernel 1: row norms for both point sets --------------------
__global__ __launch_bounds__(256) void ChamferNorms(const float* __restrict__ gts,
                                                    const float* __restrict__ preds,
                                                    float* __restrict__ norms,
                                                    int BN, int total) {
    const int row  = blockIdx.x * 8 + (threadIdx.x >> 5);
    const int lane = threadIdx.x & 31;
    if (row >= total) return;
    const float* src = (row < BN) ? (gts + (size_t)row * Dc)
                                  : (preds + (size_t)(row - BN) * Dc);
    float s = 0.0f;
    #pragma unroll
    for (int i = lane; i < Dc / 4; i += 32) {
        float4 v = ((const float4*)src)[i];
        s += v.x * v.x + v.y * v.y + v.z * v.z + v.w * v.w;
    }
    #pragma unroll
    for (int off = 16; off; off >>= 1) s += __shfl_xor(s, off, 32);
    if (lane == 0) norms[row] = s;
}

// Async-DMA one KC-wide chunk of the X and Y tiles into LDS (ASYNCcnt path).
// Each thread issues 2 b128 transfers per array: 512 float4 = 128 x 16 floats.
__device__ __forceinline__ void async_stage(const float* __restrict__ Xp,
                                            const float* __restrict__ Yp,
                                            int k0, unsigned ldsX, unsigned ldsY,
                                            int tid) {
    #pragma unroll
    for (int t = 0; t < 2; ++t) {
        const int e  = tid + t * 256;            // 0..511
        const int r  = e >> 2;                   // row 0..127
        const int c4 = (e & 3) << 2;             // col 0,4,8,12
        const unsigned goff = (unsigned)(((size_t)r * Dc + k0 + c4) * sizeof(float));
        const unsigned loff = (unsigned)((r * LDS_STR + c4) * sizeof(float));
        asm volatile("global_load_async_to_lds_b128 %0, %1, %2"
                     :: "v"(ldsX + loff), "v"(goff), "s"(Xp) : "memory");
        asm volatile("global_load_async_to_lds_b128 %0, %1, %2"
                     :: "v"(ldsY + loff), "v"(goff), "s"(Yp) : "memory");
    }
}

// ---------------- kernel 2: tiled WMMA distance + min reductions -------------
__global__ __launch_bounds__(256) void ChamferTile(const float* __restrict__ gts,
                                                   const float* __restrict__ preds,
                                                   const float* __restrict__ xx,
                                                   const float* __restrict__ yy,
                                                   unsigned* __restrict__ rowKey,
                                                   unsigned* __restrict__ colKey,
                                                   int N, int M) {
    // 4 buffers: X0 | Y0 | X1 | Y1  (double-buffered K chunks), 40 KB total
    __shared__ float lds[4 * SEG];

    const int b       = blockIdx.z;
    const int rowBase = blockIdx.y * BLK;   // gts rows
    const int colBase = blockIdx.x * BLK;   // preds cols

    const float* Xp = gts   + ((size_t)b * N + rowBase) * Dc;
    const float* Yp = preds + ((size_t)b * M + colBase) * Dc;

    const int tid     = threadIdx.x;
    const int lane    = tid & 31;
    const int wid     = tid >> 5;
    const int waveRow = wid >> 1;   // 0..3 : 32-row band
    const int waveCol = wid & 1;    // 0..1 : 64-col band
    const int half    = lane >> 4;  // hi/lo 16 lanes
    const int l16     = lane & 15;

    // LDS byte addresses (generic pointer low 32 bits == LDS offset per aperture rule)
    const unsigned ldsBase = (unsigned)(uintptr_t)(void*)&lds[0];

    v8f acc[2][4];
    #pragma unroll
    for (int i = 0; i < 2; ++i)
        #pragma unroll
        for (int j = 0; j < 4; ++j)
            #pragma unroll
            for (int e = 0; e < 8; ++e) acc[i][j][e] = 0.0f;

    // prologue: fetch chunk 0 into buffer 0
    async_stage(Xp, Yp, 0, ldsBase, ldsBase + SEG * sizeof(float), tid);
    asm volatile("s_wait_asynccnt 0x0" ::: "memory");
    __syncthreads();

    unsigned cur = 0;
    for (int c = 0; c < NCHUNK; ++c) {
        // prefetch next chunk into the other buffer while we compute this one
        if (c + 1 < NCHUNK) {
            const unsigned nb = (cur ^ 1u) * 2u * SEG * sizeof(float);
            async_stage(Xp, Yp, (c + 1) * KC,
                        ldsBase + nb, ldsBase + nb + SEG * sizeof(float), tid);
        }

        const float* Xs = lds + (size_t)cur * 2 * SEG;
        const float* Ys = Xs + SEG;

        #pragma unroll
        for (int kk = 0; kk < KC; kk += 4) {
            // A fragment (16x4 f32): lanes 0-15 at K=kk,kk+1; lanes 16-31 at K=kk+2,kk+3
            v2f afrag[2], bfrag[4];
            #pragma unroll
            for (int ti = 0; ti < 2; ++ti) {
                const int row = waveRow * 32 + ti * 16 + l16;
                afrag[ti] = *(const v2f*)(&Xs[row * LDS_STR + kk + 2 * half]);
            }
            // B fragment (4x16 f32), B[k][n] = Y[n][k] (Y^T): mirror layout
            #pragma unroll
            for (int tj = 0; tj < 4; ++tj) {
                const int col = waveCol * 64 + tj * 16 + l16;
                bfrag[tj] = *(const v2f*)(&Ys[col * LDS_STR + kk + 2 * half]);
            }
            #pragma unroll
            for (int ti = 0; ti < 2; ++ti)
                #pragma unroll
                for (int tj = 0; tj < 4; ++tj)
                    acc[ti][tj] = __builtin_amdgcn_wmma_f32_16x16x4_f32(
                        false, afrag[ti], false, bfrag[tj],
                        (short)0, acc[ti][tj], false, false);
        }

        // next chunk's async LDS writes complete, then workgroup-wide visibility
        asm volatile("s_wait_asynccnt 0x0" ::: "memory");
        __syncthreads();
        cur ^= 1u;
    }

    // ---- epilogue: d = xx + yy - 2z, row/col mins ----
    float yyv[4], xxv[2][8];
    #pragma unroll
    for (int tj = 0; tj < 4; ++tj)
        yyv[tj] = yy[(size_t)b * M + colBase + waveCol * 64 + tj * 16 + l16];
    #pragma unroll
    for (int ti = 0; ti < 2; ++ti)
        #pragma unroll
        for (int r = 0; r < 8; ++r)
            xxv[ti][r] = xx[(size_t)b * N + rowBase + waveRow * 32 + ti * 16 + half * 8 + r];

    __syncthreads();                       // done with staging buffers
    unsigned* rowmin_s = (unsigned*)lds;   // [BLK]
    unsigned* colmin_s = rowmin_s + BLK;   // [BLK]
    ((unsigned*)lds)[tid] = 0xFFFFFFFFu;   // 256 slots = both arrays
    __syncthreads();

    const float INF = __uint_as_float(0x7F800000u);
    float cpart[4];
    #pragma unroll
    for (int tj = 0; tj < 4; ++tj) cpart[tj] = INF;

    #pragma unroll
    for (int ti = 0; ti < 2; ++ti) {
        #pragma unroll
        for (int r = 0; r < 8; ++r) {
            float rpart = INF;
            #pragma unroll
            for (int tj = 0; tj < 4; ++tj) {
                const float t = -2.0f * acc[ti][tj][r];
                rpart     = fminf(rpart, yyv[tj] + t);
                cpart[tj] = fminf(cpart[tj], xxv[ti][r] + t);
            }
            atomicMin(&rowmin_s[waveRow * 32 + ti * 16 + half * 8 + r],
                      fkey(xxv[ti][r] + rpart));
        }
    }
    #pragma unroll
    for (int tj = 0; tj < 4; ++tj)
        atomicMin(&colmin_s[waveCol * 64 + tj * 16 + l16], fkey(yyv[tj] + cpart[tj]));

    __syncthreads();
    if (tid < BLK)
        atomicMin(&rowKey[(size_t)b * N + rowBase + tid], rowmin_s[tid]);
    else
        atomicMin(&colKey[(size_t)b * M + colBase + (tid - BLK)], colmin_s[tid - BLK]);
}

// ---------------- kernel 3: deterministic final sum --------------------------
__global__ __launch_bounds__(256) void ChamferFinalize(const unsigned* __restrict__ keys,
                                                       int n, float* __restrict__ out) {
    __shared__ float part[256];
    float s = 0.0f;
    for (int i = threadIdx.x; i < n; i += 256) s += fdecode(keys[i]);
    part[threadIdx.x] = s;
    __syncthreads();
    for (int w = 128; w; w >>= 1) {
        if (threadIdx.x < w) part[threadIdx.x] += part[threadIdx.x + w];
        __syncthreads();
    }
    if (threadIdx.x == 0) out[0] = part[0];
}

extern "C" void kernel_launch(void* const* d_in, const int* in_sizes, int n_in,
                              void* d_out, int out_size, void* d_ws, size_t ws_size,
                              hipStream_t stream) {
    const float* gts   = (const float*)d_in[0];
    const float* preds = (const float*)d_in[1];

    const int N = 4096, M = 4096;
    const int B  = in_sizes[0] / (N * Dc);
    const int BN = B * N, BM = B * M;

    float*    xxp    = (float*)d_ws;                 // [BN]
    float*    yyp    = xxp + BN;                     // [BM]
    unsigned* rowKey = (unsigned*)(yyp + BM);        // [BN]
    unsigned* colKey = rowKey + BN;                  // [BM] (contiguous after rowKey)

    // 0xFF fill == key 0xFFFFFFFF == +inf sentinel for the monotone key order
    hipMemsetAsync(rowKey, 0xFF, (size_t)(BN + BM) * sizeof(unsigned), stream);

    ChamferNorms<<<(BN + BM + 7) / 8, 256, 0, stream>>>(gts, preds, xxp, BN, BN + BM);

    dim3 grid(M / BLK, N / BLK, B);  // 32 x 32 x 8 = 8192 workgroups
    ChamferTile<<<grid, 256, 0, stream>>>(gts, preds, xxp, yyp, rowKey, colKey, N, M);

    ChamferFinalize<<<1, 256, 0, stream>>>(rowKey, BN + BM, (float*)d_out);
}